// YoloV1Loss_56899726737998
// MI455X (gfx1250) — compile-verified
//
#include <hip/hip_runtime.h>

// ---------------- problem constants (from reference) ----------------
#define S_GRID   7
#define NBOX     2
#define NCLS     20
#define NGT      32
#define NCELLS   (S_GRID * S_GRID)          // 49
#define FEAT     (NBOX * 5 + NCLS)          // 30
#define IMG_F    (NCELLS * FEAT)            // 1470 floats per image
#define LAMBDA_COORD 5.0f
#define LAMBDA_NOOBJ 0.5f
#define EPS_IOU  1e-6f

// ---------------- LDS layout (per block == per image) ----------------
struct __align__(16) Smem {
    float out[IMG_F + 2];        // staged predictions (pad to 16B multiple)
    float gt[NGT * 4];           // raw gt boxes (x,y,w,h)
    float gx1[NGT], gy1[NGT], gx2[NGT], gy2[NGT], garea[NGT];
    float ce[NCELLS];            // per-cell class CE
    float red[4];                // per-wave partial sums
    unsigned mask;               // 20-bit label multihot
};

// low 32 bits of a flat LDS pointer == wave-relative LDS byte offset
__device__ __forceinline__ unsigned lds_off(const void* p) {
    return (unsigned)(unsigned long long)(p);
}

// gfx1250 async DMA: global -> LDS (ASYNCcnt), GV mode (64-bit vaddr)
__device__ __forceinline__ void async_ld_b64(unsigned lds_dst, const void* g) {
    asm volatile("global_load_async_to_lds_b64 %0, %1, off"
                 :: "v"(lds_dst), "v"((unsigned long long)g) : "memory");
}
__device__ __forceinline__ void async_ld_b128(unsigned lds_dst, const void* g) {
    asm volatile("global_load_async_to_lds_b128 %0, %1, off"
                 :: "v"(lds_dst), "v"((unsigned long long)g) : "memory");
}
__device__ __forceinline__ void wait_async0() {
    asm volatile("s_wait_asynccnt 0" ::: "memory");
}

// ---------------- main kernel: one block per image ----------------
__global__ void __launch_bounds__(128)
yolo_loss_partial(const float* __restrict__ outputs,
                  const float* __restrict__ gt_boxes,
                  const int*   __restrict__ gt_labels,
                  float*       __restrict__ partial) {
    __shared__ Smem sm;
    const int tid = threadIdx.x;
    const int n   = blockIdx.x;

    // ---- stage this image's predictions via async DMA (b64 per lane) ----
    const float* gout = outputs + (size_t)n * IMG_F;
    for (int idx = tid; idx < IMG_F / 2; idx += 128)
        async_ld_b64(lds_off(&sm.out[2 * idx]), gout + 2 * idx);

    // ---- stage the 32 GT boxes (16B each, 16B aligned) via b128 DMA ----
    if (tid < NGT)
        async_ld_b128(lds_off(&sm.gt[tid * 4]),
                      gt_boxes + (size_t)n * (NGT * 4) + tid * 4);

    // ---- build label multihot while DMA is in flight ----
    int lab = 0;
    if (tid < NGT) lab = gt_labels[(size_t)n * NGT + tid];
    if (tid == 0) sm.mask = 0u;
    __syncthreads();
    if (tid < NGT) atomicOr(&sm.mask, 1u << lab);

    // drain this wave's async loads, then make LDS visible block-wide
    wait_async0();
    __syncthreads();

    // ---- precompute GT corners / areas ----
    if (tid < NGT) {
        float x = sm.gt[tid * 4 + 0], y = sm.gt[tid * 4 + 1];
        float w = sm.gt[tid * 4 + 2], h = sm.gt[tid * 4 + 3];
        sm.gx1[tid] = x - w * 0.5f;  sm.gx2[tid] = x + w * 0.5f;
        sm.gy1[tid] = y - h * 0.5f;  sm.gy2[tid] = y + h * 0.5f;
        sm.garea[tid] = w * h;
    }
    __syncthreads();

    // ---- phase A: per-cell masked cross-entropy (log-softmax) ----
    if (tid < NCELLS) {
        const float* pc = &sm.out[tid * FEAT + NBOX * 5];
        float m = pc[0];
        #pragma unroll
        for (int c = 1; c < NCLS; ++c) m = fmaxf(m, pc[c]);
        float sum = 0.f, sel = 0.f;
        const unsigned msk = sm.mask;
        #pragma unroll
        for (int c = 0; c < NCLS; ++c) {
            float v = pc[c];
            sum += expf(v - m);
            if ((msk >> c) & 1u) sel += v;
        }
        float lse = m + logf(sum);
        sm.ce[tid] = (float)__popc(msk) * lse - sel;
    }
    __syncthreads();

    // ---- phase B: per (cell, box) best-IoU match + loss terms ----
    float val = 0.f;
    if (tid < NCELLS * NBOX) {
        const int cell = tid >> 1, b = tid & 1;
        const float* pb = &sm.out[cell * FEAT + b * 5];
        const float x = pb[0], y = pb[1], w = pb[2], h = pb[3], conf = pb[4];
        const float px1 = x - w * 0.5f, px2 = x + w * 0.5f;
        const float py1 = y - h * 0.5f, py2 = y + h * 0.5f;
        const float parea = w * h;

        float best = -1.0f;
        int   bi   = 0;
        #pragma unroll 8
        for (int j = 0; j < NGT; ++j) {
            float iw = fminf(px2, sm.gx2[j]) - fmaxf(px1, sm.gx1[j]);
            float ih = fminf(py2, sm.gy2[j]) - fmaxf(py1, sm.gy1[j]);
            iw = fmaxf(iw, 0.f);
            ih = fmaxf(ih, 0.f);
            float inter = iw * ih;
            float uni   = parea + sm.garea[j] - inter;
            float iou   = inter / (uni + EPS_IOU);
            if (iou > best) { best = iou; bi = j; }   // strict '>' == argmax first-max
        }

        if (best > 0.0f) {
            float mx = sm.gt[bi * 4 + 0], my = sm.gt[bi * 4 + 1];
            float mw = sm.gt[bi * 4 + 2], mh = sm.gt[bi * 4 + 3];
            float dx = x - mx, dy = y - my;
            float dw = sqrtf(w) - sqrtf(mw);
            float dh = sqrtf(h) - sqrtf(mh);
            float dc = conf - best;
            val = LAMBDA_COORD * (dx * dx + dy * dy + dw * dw + dh * dh)
                + dc * dc + sm.ce[cell];
        } else {
            val = LAMBDA_NOOBJ * conf * conf;
        }
    }

    // ---- wave32 reduction, then 4-wave combine ----
    #pragma unroll
    for (int o = 16; o > 0; o >>= 1) val += __shfl_down(val, o, 32);
    if ((tid & 31) == 0) sm.red[tid >> 5] = val;
    __syncthreads();
    if (tid == 0)
        partial[n] = sm.red[0] + sm.red[1] + sm.red[2] + sm.red[3];
}

// ---------------- deterministic final reduction (double accumulate) ----------------
__global__ void __launch_bounds__(256)
yolo_loss_reduce(const float* __restrict__ partial, float* __restrict__ out, int nimg) {
    __shared__ double sd[256];
    const int tid = threadIdx.x;
    double acc = 0.0;
    for (int i = tid; i < nimg; i += 256) acc += (double)partial[i];
    sd[tid] = acc;
    __syncthreads();
    #pragma unroll
    for (int s = 128; s > 0; s >>= 1) {
        if (tid < s) sd[tid] += sd[tid + s];
        __syncthreads();
    }
    if (tid == 0) out[0] = (float)(sd[0] / (double)nimg);
}

// ---------------- entry point ----------------
extern "C" void kernel_launch(void* const* d_in, const int* in_sizes, int n_in,
                              void* d_out, int out_size, void* d_ws, size_t ws_size,
                              hipStream_t stream) {
    const float* outputs   = (const float*)d_in[0];
    const float* gt_boxes  = (const float*)d_in[1];
    const int*   gt_labels = (const int*)d_in[2];
    float* out     = (float*)d_out;
    float* partial = (float*)d_ws;           // nimg floats of scratch

    const int nimg = in_sizes[0] / IMG_F;    // 8192

    yolo_loss_partial<<<nimg, 128, 0, stream>>>(outputs, gt_boxes, gt_labels, partial);
    yolo_loss_reduce<<<1, 256, 0, stream>>>(partial, out, nimg);
}